// ErbEMA_52793738002703
// MI455X (gfx1250) — compile-verified
//
#include <hip/hip_runtime.h>

// ---------------------------------------------------------------------------
// ErbEMA on MI455X (gfx1250, wave32).
// Roofline: 128MiB in + 128MiB out @ 23.3 TB/s => ~11.5us; memory bound.
// 3-pass chunked linear scan (exact):
//   pass1: per-chunk zero-seeded end state A_k            (WMMA scan, no emit)
//   pass2: tiny sequential combine  S_in_k                (2048 threads)
//   pass3: seeded re-scan, emit outputs                   (WMMA scan, emit)
// Intra-chunk scan = 16x16 lower-triangular matmuls on V_WMMA_F32_16X16X4_F32:
//   NS  = M @ X + a^(t+1) (x) s_in          M[t,j] = (1-a) a^(t-j)
//   OUT = ((I-M)/40) @ X - (a^(t+1)/40) (x) s_in
// B operands are gathered straight from global memory (2x64B coalesced per
// b32 load), D tiles are stored straight back -- no LDS staging at all.
// ---------------------------------------------------------------------------

typedef __attribute__((ext_vector_type(2))) float v2f;
typedef __attribute__((ext_vector_type(8))) float v8f;

constexpr int   Bsz    = 64;      // batch
constexpr int   T      = 16384;   // time steps
constexpr int   F      = 32;      // erb bins
constexpr int   L      = 256;     // chunk length (time steps)
constexpr int   NCHUNK = T / L;   // 64 chunks per batch row
constexpr int   NBLK   = L / 16;  // 16 wmma blocks per chunk
constexpr float ALPHA  = 0.99f;
constexpr float ONE_M  = 0.01f;   // 1 - alpha
constexpr float INV40  = 0.025f;  // 1/40

template <bool EMIT>
__global__ __launch_bounds__(256) void ema_chunk_kernel(
    const float* __restrict__ x,      // [B, 1, T, F]
    float* __restrict__ out,          // [B, 1, T, F]
    float* __restrict__ carry)        // [B, NCHUNK, F]
{
    const int lane  = threadIdx.x & 31;
    const int chunk = blockIdx.x * 8 + (threadIdx.x >> 5);
    const int b     = chunk / NCHUNK;
    const int k     = chunk % NCHUNK;

    const long base_g = ((long)b * T + (long)k * L) * F;
    const float* xb = x + base_g;
    float*       ob = out + base_g;

    const int mrow    = lane & 15;
    const int khalf   = (lane >> 4) * 2;
    const int rowbase = (lane >> 4) * 8;

    // ---- constant A operands (ISA 16x4 f32 A layout:
    //      VGPR v, lane l -> element (M = l&15, K = 4s + v + 2*(l>=16))) ----
    float aM[4][2];   // M            -> new_state matmul
    float aO[4][2];   // (I-M)/40     -> output matmul
#pragma unroll
    for (int s = 0; s < 4; ++s) {
#pragma unroll
        for (int v = 0; v < 2; ++v) {
            const int kk = 4 * s + v + khalf;
            float mv = (mrow >= kk) ? ONE_M * powf(ALPHA, (float)(mrow - kk)) : 0.0f;
            aM[s][v] = mv;
            aO[s][v] = (((mrow == kk) ? 1.0f : 0.0f) - mv) * INV40;
        }
    }
    // output carry-in coefficients: D row = r + 8*(l>=16)
    float co[8];
#pragma unroll
    for (int r = 0; r < 8; ++r)
        co[r] = -powf(ALPHA, (float)(r + rowbase + 1)) * INV40;
    const float a16 = powf(ALPHA, 16.0f);   // chain decay (row 15 only is read)

    // ---- seed state replicated into D-tile column layout ----
    float sp_lo, sp_hi;   // lane l holds s[l&15] / s[16 + (l&15)]
    if (EMIT) {
        const float* cs = carry + ((long)b * NCHUNK + k) * F;
        sp_lo = cs[lane & 15];
        sp_hi = cs[16 + (lane & 15)];
    } else {
        sp_lo = 0.0f;
        sp_hi = 0.0f;
    }

    // per-lane gather base: covers rows {0,2} of each 4-row K-slice
    const float* vb = xb + (lane & 15) + 64 * (lane >> 4);

#pragma unroll
    for (int blk = 0; blk < NBLK; ++blk) {
        // ---- gather B slices straight from global (2x64B coalesced each) ----
        v2f Bs[2][4];   // [f-half][K-slice]
#pragma unroll
        for (int h = 0; h < 2; ++h) {
#pragma unroll
            for (int s = 0; s < 4; ++s) {
                Bs[h][s][0] = vb[blk * 512 + (4 * s + 0) * 32 + 16 * h];
                Bs[h][s][1] = vb[blk * 512 + (4 * s + 1) * 32 + 16 * h];
            }
        }

#pragma unroll
        for (int h = 0; h < 2; ++h) {
            // NS = M @ X (chained K=4 f32 WMMAs)
            v8f ns = {};
#pragma unroll
            for (int s = 0; s < 4; ++s) {
                v2f Am; Am[0] = aM[s][0]; Am[1] = aM[s][1];
                ns = __builtin_amdgcn_wmma_f32_16x16x4_f32(
                        false, Am, false, Bs[h][s], (short)0, ns, false, false);
            }

            const float sp = h ? sp_hi : sp_lo;

            if (EMIT) {
                // OUT = ((I-M)/40) @ X - (a^(t+1)/40) * sp
                v8f ot = {};
#pragma unroll
                for (int s = 0; s < 4; ++s) {
                    v2f Ao; Ao[0] = aO[s][0]; Ao[1] = aO[s][1];
                    ot = __builtin_amdgcn_wmma_f32_16x16x4_f32(
                            false, Ao, false, Bs[h][s], (short)0, ot, false, false);
                }
                float* dst = ob + (long)(blk * 16 + rowbase) * F + (lane & 15) + 16 * h;
#pragma unroll
                for (int r = 0; r < 8; ++r)
                    dst[r * F] = fmaf(co[r], sp, ot[r]);   // 2x64B coalesced per store
            }

            // carry chain: s' = NS row15 + a^16*sp; row15 = v7, lanes 16..31
            float v7  = fmaf(a16, sp, ns[7]);
            float nsp = __int_as_float(__builtin_amdgcn_ds_bpermute(
                            (16 + (lane & 15)) * 4, __float_as_int(v7)));
            if (h == 0) sp_lo = nsp; else sp_hi = nsp;
        }
    }

    if (!EMIT) {
        // A_k[f]: lanes 0..15 carry sp_lo == A[lane]; lanes 16..31 sp_hi == A[lane]
        float* cdst = carry + ((long)b * NCHUNK + k) * F;
        cdst[lane] = (lane < 16) ? sp_lo : sp_hi;
    }
}

// ---------------------------------------------------------------------------
// Pass 2: sequential combine across chunks (tiny). In-place:
//   carry[b][k][f] : A_k -> S_in_k ;  final_state[b][f] = S_in_NCHUNK
// ---------------------------------------------------------------------------
__global__ __launch_bounds__(256) void ema_carry_scan_kernel(
    const float* __restrict__ state0,   // [1,1,F]
    float* __restrict__ carry,          // [B, NCHUNK, F]
    float* __restrict__ final_state)    // [B, F]
{
    const int tid = blockIdx.x * blockDim.x + threadIdx.x;
    if (tid >= Bsz * F) return;
    const int b = tid >> 5;
    const int f = tid & 31;

    const float alphaL = powf(ALPHA, (float)L);
    float S = state0[f];
    float* c = carry + ((long)b * NCHUNK) * F + f;
#pragma unroll 4
    for (int kk = 0; kk < NCHUNK; ++kk) {
        float A = c[(long)kk * F];
        c[(long)kk * F] = S;            // incoming state for chunk kk
        S = fmaf(alphaL, S, A);         // end state of chunk kk
    }
    final_state[b * F + f] = S;
}

// ---------------------------------------------------------------------------
extern "C" void kernel_launch(void* const* d_in, const int* in_sizes, int n_in,
                              void* d_out, int out_size, void* d_ws, size_t ws_size,
                              hipStream_t stream) {
    const float* x      = (const float*)d_in[0];   // feat_erb [64,1,16384,32] f32
    const float* state0 = (const float*)d_in[1];   // state    [1,1,32]       f32
    float* out    = (float*)d_out;                          // feat_out, then final_state
    float* fstate = out + (long)Bsz * T * F;                // [64,1,32]
    float* carry  = (float*)d_ws;                           // B*NCHUNK*F f32 = 512 KiB

    const int nchunks = Bsz * NCHUNK;            // 4096 independent waves
    ema_chunk_kernel<false><<<nchunks / 8, 256, 0, stream>>>(x, out, carry);
    ema_carry_scan_kernel<<<(Bsz * F + 255) / 256, 256, 0, stream>>>(state0, carry, fstate);
    ema_chunk_kernel<true><<<nchunks / 8, 256, 0, stream>>>(x, out, carry);
}